// QuantumClassifier_9577777070217
// MI455X (gfx1250) — compile-verified
//
#include <hip/hip_runtime.h>
#include <math.h>

// ---------------------------------------------------------------------------
// Quantum circuit classifier on gfx1250 (MI455X), wave32 + V_WMMA_F32_16X16X4_F32
//
// State per batch element: 1024 complex amps viewed as 32x32 complex matrix S,
//   flat index a[9:0]:  row r = a[9:5] (qubits 0-4), col c = a[4:0] (qubits 5-9).
// Per layer:
//   S <- U1hi * S          (fused RZ@RX gates, qubits 0-4)   [row matmul]
//   S <- S * U1lo^T        (fused RZ@RX gates, qubits 5-9)   [col matmul]
//   S <- Perm(S)           (fixed CNOT network, index permutation)
//   S <- U2hi * S          (fused RX@RZ gates, qubits 0-4)
//   S <- S * U2lo^T        (fused RX@RZ gates, qubits 5-9)
// Complex matmul via 4 real matmuls -> v_wmma_f32_16x16x4_f32 tiles.
// LDS arrays padded to 34-float row stride: conflict-free A loads/C stores
// across half-wave lane groups (64 banks), and 8B alignment for b64 K-pairs.
// k-outer matmul loop: A/B tiles loaded once per k-step, 16 WMMAs issued.
// ---------------------------------------------------------------------------

#define NQ     10
#define DIM    1024
#define NLAY   10
#define STR    34                    // padded row stride (floats, even)
#define MSZ    (32 * STR)            // padded 32x32 matrix size (1088 floats)

typedef float v2f __attribute__((ext_vector_type(2)));
typedef float v8f __attribute__((ext_vector_type(8)));

__device__ __forceinline__ v8f wmma4(v2f a, v2f b, v8f c) {
  // 8-arg pattern: (neg_a, A, neg_b, B, c_mod, C, reuse_a, reuse_b)
  return __builtin_amdgcn_wmma_f32_16x16x4_f32(false, a, false, b, (short)0, c,
                                               false, false);
}

__device__ __forceinline__ v2f ld2(const float* p) {   // 8B-aligned pair -> b64
  return *(const v2f*)p;
}

__device__ __forceinline__ float wave_sum(float v) {
#pragma unroll
  for (int m = 16; m >= 1; m >>= 1) v += __shfl_xor(v, m, 32);
  return v;
}

// D = U * S   (complex 32x32). ur/ui/un(= -ui): stride-34. sr/si: state.
__device__ __forceinline__ void mm_row(const float* __restrict__ ur,
                                       const float* __restrict__ ui,
                                       const float* __restrict__ un,
                                       float* __restrict__ sr,
                                       float* __restrict__ si, int lane) {
  const int ml = lane & 15;          // A: M / B: N lane index
  const int kk = (lane >> 4) << 1;   // K pair base (0 or 2)
  v8f accR[4], accI[4];
#pragma unroll
  for (int t = 0; t < 4; ++t) { accR[t] = (v8f)(0.f); accI[t] = (v8f)(0.f); }

  const int abase = ml * STR + kk;   // lane-dependent part of A offsets (even)
#pragma unroll
  for (int k8 = 0; k8 < 8; ++k8) {
    const int kb = k8 * 4;
    v2f Ar[2], Ai[2], An[2], Br[2], Bi[2];
#pragma unroll
    for (int ti = 0; ti < 2; ++ti) {
      const int ao = abase + ti * (16 * STR) + kb;   // A[m][k]=U[ti16+m][kb+k]
      Ar[ti] = ld2(ur + ao);
      Ai[ti] = ld2(ui + ao);
      An[ti] = ld2(un + ao);
    }
#pragma unroll
    for (int tj = 0; tj < 2; ++tj) {
      const int bo = (kb + kk) * STR + tj * 16 + ml; // B[k][n]=S[kb+k][tj16+n]
      const float* bpr = sr + bo;
      const float* bpi = si + bo;
      Br[tj] = v2f{ bpr[0], bpr[STR] };
      Bi[tj] = v2f{ bpi[0], bpi[STR] };
    }
#pragma unroll
    for (int ti = 0; ti < 2; ++ti)
#pragma unroll
      for (int tj = 0; tj < 2; ++tj) {
        const int t = ti * 2 + tj;
        accR[t] = wmma4(Ar[ti], Br[tj], accR[t]);
        accR[t] = wmma4(An[ti], Bi[tj], accR[t]);   // - Ui*Si
        accI[t] = wmma4(Ar[ti], Bi[tj], accI[t]);
        accI[t] = wmma4(Ai[ti], Br[tj], accI[t]);
      }
  }
  const int rowoff = (lane >> 4) << 3;  // C: lanes 16-31 hold rows +8
#pragma unroll
  for (int ti = 0; ti < 2; ++ti)
#pragma unroll
    for (int tj = 0; tj < 2; ++tj) {
      v8f r = accR[ti * 2 + tj], q = accI[ti * 2 + tj];
#pragma unroll
      for (int i = 0; i < 8; ++i) {
        const int idx = (ti * 16 + rowoff + i) * STR + tj * 16 + ml;
        sr[idx] = r[i]; si[idx] = q[i];
      }
    }
}

// D = S * U^T (complex 32x32): Dr = Sr*Ur^T - Si*Ui^T ; Di = Sr*Ui^T + Si*Ur^T
__device__ __forceinline__ void mm_col(const float* __restrict__ ur,
                                       const float* __restrict__ ui,
                                       const float* __restrict__ un,
                                       float* __restrict__ sr,
                                       float* __restrict__ si, int lane) {
  const int ml = lane & 15;
  const int kk = (lane >> 4) << 1;
  v8f accR[4], accI[4];
#pragma unroll
  for (int t = 0; t < 4; ++t) { accR[t] = (v8f)(0.f); accI[t] = (v8f)(0.f); }

  const int base = ml * STR + kk;    // shared lane-dependent offset (even)
#pragma unroll
  for (int k8 = 0; k8 < 8; ++k8) {
    const int kb = k8 * 4;
    v2f Asr[2], Asi[2], Bur[2], Bui[2], Bun[2];
#pragma unroll
    for (int ti = 0; ti < 2; ++ti) {
      const int ao = base + ti * (16 * STR) + kb;    // A[m][k]=S[ti16+m][kb+k]
      Asr[ti] = ld2(sr + ao);
      Asi[ti] = ld2(si + ao);
    }
#pragma unroll
    for (int tj = 0; tj < 2; ++tj) {
      const int bo = base + tj * (16 * STR) + kb;    // B[k][n]=U[tj16+n][kb+k]
      Bur[tj] = ld2(ur + bo);
      Bui[tj] = ld2(ui + bo);
      Bun[tj] = ld2(un + bo);
    }
#pragma unroll
    for (int ti = 0; ti < 2; ++ti)
#pragma unroll
      for (int tj = 0; tj < 2; ++tj) {
        const int t = ti * 2 + tj;
        accR[t] = wmma4(Asr[ti], Bur[tj], accR[t]);
        accR[t] = wmma4(Asi[ti], Bun[tj], accR[t]);  // - Si*Ui^T
        accI[t] = wmma4(Asr[ti], Bui[tj], accI[t]);
        accI[t] = wmma4(Asi[ti], Bur[tj], accI[t]);
      }
  }
  const int rowoff = (lane >> 4) << 3;
#pragma unroll
  for (int ti = 0; ti < 2; ++ti)
#pragma unroll
    for (int tj = 0; tj < 2; ++tj) {
      v8f r = accR[ti * 2 + tj], q = accI[ti * 2 + tj];
#pragma unroll
      for (int i = 0; i < 8; ++i) {
        const int idx = (ti * 16 + rowoff + i) * STR + tj * 16 + ml;
        sr[idx] = r[i]; si[idx] = q[i];
      }
    }
}

// Dynamic LDS layout (floats):
//   U   : 12 * MSZ    (4 matrices x {Ur, Ui, -Ui}, stride-34)
//   S   : 16 * MSZ    (8 waves x {Sr, Si}, stride-34)
//   G   : 160         (2 kinds x 10 qubits x 2x2 complex)
//   perm: 1024 u16    (512 float slots) -- holds PADDED source index
#define SMEM_FLOATS (12 * MSZ + 16 * MSZ + 160 + 512)

__global__ __launch_bounds__(256) void
qsim_kernel(const float* __restrict__ x, const float* __restrict__ w,
            const float* __restrict__ Wl, const float* __restrict__ bl,
            float* __restrict__ out, int nElem) {
  extern __shared__ float smem[];
  float* U = smem;                       // [12][MSZ]
  float* S = smem + 12 * MSZ;            // [16][MSZ]
  float* G = smem + 28 * MSZ;            // [160]
  unsigned short* perm = (unsigned short*)(G + 160);

  const int tid  = threadIdx.x;
  const int wave = tid >> 5;
  const int lane = tid & 31;
  const int e    = blockIdx.x * 8 + wave;
  const bool live = (e < nElem);

  float* sr = S + (2 * wave) * MSZ;
  float* si = sr + MSZ;

  // ---- build fixed CNOT permutation table (padded source index) ----
  {
    const int seq[9] = {1, 3, 5, 7, 0, 2, 4, 6, 8};  // control qubits, tgt=c+1
    for (int idx = tid; idx < 1024; idx += 256) {
      int p = idx;
      // new[a] = old[f(a)], f = m1( m2( ... m18(a) ) ): apply gates in reverse
#pragma unroll
      for (int rep = 0; rep < 2; ++rep)
#pragma unroll
        for (int gI = 8; gI >= 0; --gI) {
          const int cq = seq[gI];
          const int cb = 9 - cq, tb = 8 - cq;   // bit of control / target qubit
          p ^= ((p >> cb) & 1) << tb;
        }
      perm[idx] = (unsigned short)((p >> 5) * STR + (p & 31));
    }
  }

  // ---- load + preprocess input state (amplitude embedding) ----
  {
    const float* xp = x + (size_t)(live ? e : 0) * DIM;
    float v[32];
    float ss = 0.f;
#pragma unroll
    for (int i = 0; i < 32; ++i) {
      float f = xp[i * 32 + lane];
      if (__builtin_isnan(f)) f = 0.f;
      else if (__builtin_isinf(f)) f = (f > 0.f) ? 1.f : -1.f;
      f += 1e-8f;
      v[i] = f;
      ss += f * f;
    }
    ss = wave_sum(ss);
    float sc = 1.f / fmaxf(sqrtf(ss), 1e-12f);
    ss = 0.f;
#pragma unroll
    for (int i = 0; i < 32; ++i) {
      float f = v[i] * sc;
      f = fminf(1.f, fmaxf(-1.f, f));
      v[i] = f;
      ss += f * f;
    }
    ss = wave_sum(ss);
    sc = 1.f / sqrtf(ss);
#pragma unroll
    for (int i = 0; i < 32; ++i) {
      sr[i * STR + lane] = v[i] * sc;    // amp index a = i*32 + lane
      si[i * STR + lane] = 0.f;
    }
  }

  // ---- layers ----
  for (int layer = 0; layer < NLAY; ++layer) {
    __syncthreads();                      // previous layer done reading U

    // 2x2 gates: kind0 = RZ@RX (pre), kind1 = RX@RZ (post)
    if (tid < NQ) {
      const float a  = w[(layer * NQ + tid) * 2 + 0];
      const float th = w[(layer * NQ + tid) * 2 + 1];
      const float ca = cosf(0.5f * a),  sa = sinf(0.5f * a);
      const float cb = cosf(0.5f * th), sb = sinf(0.5f * th);
      float* g0 = G + ((0 * NQ + tid) * 4) * 2;   // [r][c] re,im interleaved
      float* g1 = G + ((1 * NQ + tid) * 4) * 2;
      // G1 = RZ@RX
      g0[0] =  cb * ca; g0[1] = -cb * sa;      // [0][0]
      g0[2] = -sb * sa; g0[3] = -sb * ca;      // [0][1]
      g0[4] =  sb * sa; g0[5] = -sb * ca;      // [1][0]
      g0[6] =  cb * ca; g0[7] =  cb * sa;      // [1][1]
      // G2 = RX@RZ
      g1[0] =  cb * ca; g1[1] = -cb * sa;
      g1[2] =  sb * sa; g1[3] = -sb * ca;
      g1[4] = -sb * sa; g1[5] = -sb * ca;
      g1[6] =  cb * ca; g1[7] =  cb * sa;
    }
    __syncthreads();

    // Build fused 32x32 Kronecker unitaries: m = {U1hi, U1lo, U2hi, U2lo}
#pragma unroll
    for (int m = 0; m < 4; ++m) {
      const int kind = m >> 1, qb = (m & 1) * 5;
      float* ur = U + (3 * m + 0) * MSZ;
      float* ui = U + (3 * m + 1) * MSZ;
      float* un = U + (3 * m + 2) * MSZ;
      for (int idx = tid; idx < 1024; idx += 256) {
        const int r = idx >> 5, c = idx & 31;
        float re = 1.f, im = 0.f;
#pragma unroll
        for (int k = 0; k < 5; ++k) {
          const int rb = (r >> (4 - k)) & 1, cbit = (c >> (4 - k)) & 1;
          const float* g = G + (((kind * NQ + qb + k) * 2 + rb) * 2 + cbit) * 2;
          const float gr = g[0], gi = g[1];
          const float nre = re * gr - im * gi;
          im = re * gi + im * gr;
          re = nre;
        }
        const int o = r * STR + c;
        ur[o] = re; ui[o] = im; un[o] = -im;
      }
    }
    __syncthreads();

    // Apply: rows, cols, CNOT perm, rows, cols (per-wave, own LDS state)
    mm_row(U + 0 * 3 * MSZ, U + (0 * 3 + 1) * MSZ, U + (0 * 3 + 2) * MSZ, sr, si, lane);
    mm_col(U + 1 * 3 * MSZ, U + (1 * 3 + 1) * MSZ, U + (1 * 3 + 2) * MSZ, sr, si, lane);

    {
      float tr[32], tq[32];
#pragma unroll
      for (int i = 0; i < 32; ++i) {
        const int p = perm[i * 32 + lane];   // padded index
        tr[i] = sr[p]; tq[i] = si[p];
      }
#pragma unroll
      for (int i = 0; i < 32; ++i) {
        sr[i * STR + lane] = tr[i];
        si[i * STR + lane] = tq[i];
      }
    }

    mm_row(U + 2 * 3 * MSZ, U + (2 * 3 + 1) * MSZ, U + (2 * 3 + 2) * MSZ, sr, si, lane);
    mm_col(U + 3 * 3 * MSZ, U + (3 * 3 + 1) * MSZ, U + (3 * 3 + 2) * MSZ, sr, si, lane);
  }

  // ---- readout: <Z0> over bit9 (= row MSB), then Linear(1,10) + log_softmax
  float acc = 0.f;
#pragma unroll
  for (int r = 0; r < 32; ++r) {
    const float re = sr[r * STR + lane], im = si[r * STR + lane];
    const float p2 = re * re + im * im;
    acc += (r < 16) ? p2 : -p2;
  }
  const float ev = wave_sum(acc);

  if (lane == 0 && live) {
    float l[10];
    float mx = -INFINITY;
#pragma unroll
    for (int j = 0; j < 10; ++j) {
      l[j] = ev * Wl[j] + bl[j];
      mx = fmaxf(mx, l[j]);
    }
    float s = 0.f;
#pragma unroll
    for (int j = 0; j < 10; ++j) s += expf(l[j] - mx);
    const float lse = logf(s);
#pragma unroll
    for (int j = 0; j < 10; ++j) out[(size_t)e * 10 + j] = l[j] - mx - lse;
  }
}

extern "C" void kernel_launch(void* const* d_in, const int* in_sizes, int n_in,
                              void* d_out, int out_size, void* d_ws, size_t ws_size,
                              hipStream_t stream) {
  (void)n_in; (void)d_ws; (void)ws_size; (void)out_size;
  const float* x  = (const float*)d_in[0];   // [B, 1024]
  const float* w  = (const float*)d_in[1];   // [10, 10, 2]
  const float* Wl = (const float*)d_in[2];   // [10, 1]
  const float* bl = (const float*)d_in[3];   // [10]
  float* out = (float*)d_out;                // [B, 10]

  const int nElem = in_sizes[0] / DIM;       // 4096
  const int blocks = (nElem + 7) / 8;        // 8 waves (elements) per block
  const size_t smem = (size_t)SMEM_FLOATS * sizeof(float);

  qsim_kernel<<<blocks, 256, smem, stream>>>(x, w, Wl, bl, out, nElem);
}